// Filtration_23665269801453
// MI455X (gfx1250) — compile-verified
//
#include <hip/hip_runtime.h>
#include <math.h>

typedef float v2f __attribute__((ext_vector_type(2)));
typedef float v8f __attribute__((ext_vector_type(8)));

// ---------------------------------------------------------------------------
// Build x0 = concat(emb_deg[node_deg], emb_lab[node_lab]) into XC cols [0,128)
// XC row stride = 320 floats. One thread per float4 (N * 32 threads).
// ---------------------------------------------------------------------------
__global__ __launch_bounds__(256) void build_x0_kernel(
    const int* __restrict__ deg, const int* __restrict__ lab,
    const float* __restrict__ emb_deg, const float* __restrict__ emb_lab,
    float* __restrict__ XC, int N)
{
    int tid = blockIdx.x * blockDim.x + threadIdx.x;
    int n = tid >> 5;
    if (n >= N) return;
    int c4 = (tid & 31) << 2;        // 0..124 step 4
    float4 v;
    if (c4 < 64)
        v = *(const float4*)(emb_deg + (size_t)deg[n] * 64 + c4);
    else
        v = *(const float4*)(emb_lab + (size_t)lab[n] * 64 + (c4 - 64));
    *(float4*)(XC + (size_t)n * 320 + c4) = v;
}

// ---------------------------------------------------------------------------
// Zero fill
// ---------------------------------------------------------------------------
__global__ __launch_bounds__(256) void fillz_kernel(float* __restrict__ p, long long n)
{
    long long i = (long long)blockIdx.x * blockDim.x + threadIdx.x;
    if (i < n) p[i] = 0.f;
}

// ---------------------------------------------------------------------------
// Edge scatter: AGG[dst][c:c+4] += X[src][xoff+c : xoff+c+4]
// One thread per (edge, 4-column chunk); chunks per edge = 1<<shift.
// Consecutive threads -> same edge, consecutive chunks (coalesced).
// ---------------------------------------------------------------------------
__global__ __launch_bounds__(256) void scatter_kernel(
    const int* __restrict__ ei,           // [2, E]
    const float* __restrict__ X, int ldx, int xoff,
    float* __restrict__ AGG, int kagg, int shift, int E)
{
    int tid = blockIdx.x * blockDim.x + threadIdx.x;
    int total = E << shift;
    if (tid >= total) return;
    int e = tid >> shift;
    int c = (tid - (e << shift)) << 2;
    int src = ei[e];
    int dst = ei[E + e];
    const float4 v = *(const float4*)(X + (size_t)src * ldx + xoff + c);
    float* outp = AGG + (size_t)dst * kagg + c;
    atomicAdd(outp + 0, v.x);
    atomicAdd(outp + 1, v.y);
    atomicAdd(outp + 2, v.z);
    atomicAdd(outp + 3, v.w);
}

// ---------------------------------------------------------------------------
// Fused WMMA GEMM:  H[N,64] = (X[:, xoff:xoff+K] (+ AGG[N,K])) @ W[64,K]^T + b
// plus per-column sum / sum-of-squares accumulation for BatchNorm.
//
// One wave (32 lanes) per 16x16 output tile; each wave walks `tiles` row
// tiles so BN atomics amortize. Uses V_WMMA_F32_16X16X4_F32.
//
// VGPR layouts per CDNA5 ISA 7.12.2 (f32, wave32):
//   A 16x4 : lane = half*16 + lr ; a[j] = A[M=lr][K = 2*half + j]
//   B 4x16 : b[j] = B[K = 2*half + j][N = lr] = W[n0+lr][k + 2*half + j]
//   C 16x16: acc[i] @ lane -> row M = i + 8*half, col N = lr
// ---------------------------------------------------------------------------
template <bool HAS_AGG>
__global__ __launch_bounds__(32) void gemm_wmma_kernel(
    const float* __restrict__ X, int ldx, int xoff,
    const float* __restrict__ AGG,        // [N, K] dense (ignored if !HAS_AGG)
    const float* __restrict__ W,          // [64, K]
    const float* __restrict__ bias,       // [64]
    float* __restrict__ H,                // [N, 64]
    float* __restrict__ stats,            // [128]: col sums, col sumsq
    int K, int numRowTiles, int tiles)
{
    const int lane = threadIdx.x;
    const int half = lane >> 4;
    const int lr   = lane & 15;
    const int n0   = blockIdx.y << 4;     // output column tile base
    const float bcol = bias[n0 + lr];
    const float* wrow = W + (size_t)(n0 + lr) * K + 2 * half;

    float csum = 0.f, csumsq = 0.f;
    const int rt0 = blockIdx.x * tiles;

    for (int rt = rt0; rt < rt0 + tiles; ++rt) {
        if (rt >= numRowTiles) break;
        const int m0 = rt << 4;
        const float* xrow = X + (size_t)(m0 + lr) * ldx + xoff + 2 * half;
        const float* grow = HAS_AGG ? (AGG + (size_t)(m0 + lr) * K + 2 * half)
                                    : nullptr;
        v8f acc = {0.f, 0.f, 0.f, 0.f, 0.f, 0.f, 0.f, 0.f};

        for (int k = 0; k < K; k += 4) {
            v2f a, b;
            a.x = xrow[k];
            a.y = xrow[k + 1];
            if (HAS_AGG) {
                a.x += grow[k];
                a.y += grow[k + 1];
            }
            b.x = wrow[k];
            b.y = wrow[k + 1];
            acc = __builtin_amdgcn_wmma_f32_16x16x4_f32(
                false, a, false, b, (short)0, acc, false, false);
        }

        float* hout = H + (size_t)m0 * 64 + (n0 + lr);
#pragma unroll
        for (int i = 0; i < 8; ++i) {
            float hv = acc[i] + bcol;
            int row = i + 8 * half;
            hout[(size_t)row * 64] = hv;
            csum += hv;
            csumsq += hv * hv;
        }
    }

    // lanes L and L+16 hold the same output column n0+lr: merge, then one
    // atomic per column from the low half.
    csum   += __shfl_xor(csum, 16, 32);
    csumsq += __shfl_xor(csumsq, 16, 32);
    if (half == 0) {
        atomicAdd(&stats[n0 + lr], csum);
        atomicAdd(&stats[64 + n0 + lr], csumsq);
    }
}

// ---------------------------------------------------------------------------
// BatchNorm (training stats from sums) + LeakyReLU(0.01).
// One thread per (node, 4 columns): N*16 threads.
// ---------------------------------------------------------------------------
__global__ __launch_bounds__(256) void bn_act_kernel(
    const float* __restrict__ H,          // [N, 64]
    const float* __restrict__ stats,      // [128]
    const float* __restrict__ g, const float* __restrict__ be,
    float* __restrict__ Z, int ldz, int zoff, int N, float invN)
{
    int tid = blockIdx.x * blockDim.x + threadIdx.x;
    if (tid >= N * 16) return;
    int n = tid >> 4;
    int c = (tid & 15) << 2;
    float4 h = *(const float4*)(H + (size_t)n * 64 + c);
    float hv[4] = {h.x, h.y, h.z, h.w};
    float o[4];
#pragma unroll
    for (int j = 0; j < 4; ++j) {
        float m   = stats[c + j] * invN;
        float var = stats[64 + c + j] * invN - m * m;
        float s   = rsqrtf(var + 1e-5f);
        float t   = (hv[j] - m) * s * g[c + j] + be[c + j];
        o[j] = (t >= 0.f) ? t : 0.01f * t;
    }
    float4 ov = {o[0], o[1], o[2], o[3]};
    *(float4*)(Z + (size_t)n * ldz + zoff + c) = ov;
}

// ---------------------------------------------------------------------------
// out[n] = sigmoid( dot(ZF[n,:], fw2) + fb2 )
// ---------------------------------------------------------------------------
__global__ __launch_bounds__(256) void final_out_kernel(
    const float* __restrict__ ZF, const float* __restrict__ fw2,
    const float* __restrict__ fb2, float* __restrict__ out, int N)
{
    int n = blockIdx.x * blockDim.x + threadIdx.x;
    if (n >= N) return;
    float s = fb2[0];
    const float4* z = (const float4*)(ZF + (size_t)n * 64);
#pragma unroll
    for (int i = 0; i < 16; ++i) {
        float4 zv = z[i];
        float4 wv = *(const float4*)(fw2 + i * 4);
        s += zv.x * wv.x + zv.y * wv.y + zv.z * wv.z + zv.w * wv.w;
    }
    out[n] = 1.f / (1.f + expf(-s));
}

// ---------------------------------------------------------------------------
extern "C" void kernel_launch(void* const* d_in, const int* in_sizes, int n_in,
                              void* d_out, int out_size, void* d_ws, size_t ws_size,
                              hipStream_t stream)
{
    const int N = in_sizes[0];
    const int E = in_sizes[2] / 2;

    const int*   node_deg = (const int*)d_in[0];
    const int*   node_lab = (const int*)d_in[1];
    const int*   ei       = (const int*)d_in[2];
    const float* emb_deg  = (const float*)d_in[3];
    const float* emb_lab  = (const float*)d_in[4];
    const float* w[3]  = {(const float*)d_in[5],  (const float*)d_in[9],  (const float*)d_in[13]};
    const float* bb[3] = {(const float*)d_in[6],  (const float*)d_in[10], (const float*)d_in[14]};
    const float* gg[3] = {(const float*)d_in[7],  (const float*)d_in[11], (const float*)d_in[15]};
    const float* bei[3]= {(const float*)d_in[8],  (const float*)d_in[12], (const float*)d_in[16]};
    const float* fw1 = (const float*)d_in[17];
    const float* fb1 = (const float*)d_in[18];
    const float* fg  = (const float*)d_in[19];
    const float* fbe = (const float*)d_in[20];
    const float* fw2 = (const float*)d_in[21];
    const float* fb2 = (const float*)d_in[22];

    // Workspace layout (floats)
    float* XC   = (float*)d_ws;                    // [N, 320] persistent concat
    float* AGG  = XC  + (size_t)N * 320;           // [N, 128] scatter scratch
    float* H    = AGG + (size_t)N * 128;           // [N, 64]  pre-BN activations
    float* ZF   = H   + (size_t)N * 64;            // [N, 64]  final hidden
    float* stats= ZF  + (size_t)N * 64;            // [128]    colsum / colsumsq

    const float invN = 1.0f / (float)N;

    // 1) input embedding -> XC[:, 0:128]
    {
        int total = N * 32;
        build_x0_kernel<<<(total + 255) / 256, 256, 0, stream>>>(
            node_deg, node_lab, emb_deg, emb_lab, XC, N);
    }

    const int numRowTiles = (N + 15) / 16;          // 6250 for N=100000
    const int tilesPerWave = 5;
    const int gemmGx = (numRowTiles + tilesPerWave - 1) / tilesPerWave;

    const int inOff[3] = {0, 128, 192};
    const int Kin[3]   = {128, 64, 64};
    const int zoff[3]  = {128, 192, 256};

    // 2) three GIN layers
    for (int i = 0; i < 3; ++i) {
        long long aggN = (long long)N * Kin[i];
        fillz_kernel<<<(int)((aggN + 255) / 256), 256, 0, stream>>>(AGG, aggN);
        fillz_kernel<<<1, 128, 0, stream>>>(stats, 128);

        int shift = (Kin[i] == 128) ? 5 : 4;        // float4 chunks per edge
        int total = E << shift;
        scatter_kernel<<<(total + 255) / 256, 256, 0, stream>>>(
            ei, XC, 320, inOff[i], AGG, Kin[i], shift, E);

        gemm_wmma_kernel<true><<<dim3(gemmGx, 4), 32, 0, stream>>>(
            XC, 320, inOff[i], AGG, w[i], bb[i], H, stats,
            Kin[i], numRowTiles, tilesPerWave);

        bn_act_kernel<<<(N * 16 + 255) / 256, 256, 0, stream>>>(
            H, stats, gg[i], bei[i], XC, 320, zoff[i], N, invN);
    }

    // 3) final MLP head: [N,320] @ fw1^T -> BN -> leaky -> dot fw2 -> sigmoid
    fillz_kernel<<<1, 128, 0, stream>>>(stats, 128);
    gemm_wmma_kernel<false><<<dim3(gemmGx, 4), 32, 0, stream>>>(
        XC, 320, 0, nullptr, fw1, fb1, H, stats, 320, numRowTiles, tilesPerWave);
    bn_act_kernel<<<(N * 16 + 255) / 256, 256, 0, stream>>>(
        H, stats, fg, fbe, ZF, 64, 0, N, invN);
    final_out_kernel<<<(N + 255) / 256, 256, 0, stream>>>(
        ZF, fw2, fb2, (float*)d_out, N);
}